// Gen2DMap_53429393162557
// MI455X (gfx1250) — compile-verified
//
#include <hip/hip_runtime.h>
#include <stdint.h>

// Problem constants (from the reference): x[32, 64, 512] f32 -> out[32, 512, 64, 64] f32.
#define MAP_N   64
#define DMODEL  512
#define BATCH   32
#define D_TILE  16          // channels per workgroup (one wave32 per channel on output)
#define NLEV    7           // sparse-table levels: windows 1,2,4,...,64
#define RPAD    65          // 64 + 1 pad: stride-65 rows -> conflict-free LDS banks
#define NTHR    512         // 16 waves

typedef __attribute__((ext_vector_type(4))) float v4f;
typedef __attribute__((address_space(1))) int as1_int;
typedef __attribute__((address_space(3))) int as3_int;

// out(r,c) for one channel tile: range-max via sparse table, zero off the mask.
__device__ __forceinline__ float tile_val(const float (&M)[NLEV][D_TILE][RPAD],
                                          int dl, int r, int c) {
  const int t = c - r;
  const unsigned ut = (unsigned)t;          // negative t -> huge -> all groups false
  // group1: t in [0,15], any r.  group2: odd t in [17,31], even r.
  // group3: t%4==3 in [35,63], r%4==0.  (t&3)==3 already excludes t in [32,34].
  const int valid = (ut <= 15u)
                  | ((ut <= 31u) & ((t & 1) != 0) & ((r & 1) == 0))
                  | ((ut <= 63u) & ((t & 3) == 3) & ((r & 3) == 0));
  // k = floor(log2(t+1)), clamped into [0,6] so speculative addressing stays in range.
  const int k = 31 - __builtin_clz((ut & 63u) + 1u);
  if (!valid) return 0.0f;
  const float a = M[k][dl][r];                    // covers [r, r+2^k-1] (clamped) ⊆ [r,c]
  const float b = M[k][dl][c + 1 - (1 << k)];     // covers [c-2^k+1, c]
  return fmaxf(a, b);
}

extern "C" __global__ void __launch_bounds__(NTHR, 2)
gen2dmap_kernel(const float* __restrict__ x, float* __restrict__ out) {
  __shared__ float M[NLEV][D_TILE][RPAD];

  const int tid = threadIdx.x;
  const int b   = blockIdx.x >> 5;              // 32 batches
  const int d0  = (blockIdx.x & 31) * D_TILE;   // 32 channel groups of 16

  // ---- Phase 1: async global -> LDS stage of x[b, 0:64, d0:d0+16] into level 0 ----
  // element (n, dc):  x[(b*64 + n)*512 + d0 + dc]  ->  M[0][dc][n]
  for (int e = tid; e < D_TILE * MAP_N; e += NTHR) {   // 1024 elems, 2 per thread
    const int dc = e & (D_TILE - 1);
    const int n  = e >> 4;
    const float* gp = x + ((size_t)(b * MAP_N + n) * DMODEL + (d0 + dc));
    float* lp = &M[0][dc][n];
#if defined(__gfx1250__) && __has_builtin(__builtin_amdgcn_global_load_async_to_lds_b32)
    __builtin_amdgcn_global_load_async_to_lds_b32(
        (as1_int*)(uintptr_t)gp,   // global src (cast away const via integer)
        (as3_int*)lp,              // LDS dst
        /*offset=*/0, /*cpol=*/0);
#else
    const unsigned lds_off = (unsigned)(uintptr_t)lp;  // addr[31:0] = LDS byte offset
    asm volatile("global_load_async_to_lds_b32 %0, %1, off"
                 :: "v"(lds_off), "v"(gp) : "memory");
#endif
  }
#if defined(__gfx1250__) && __has_builtin(__builtin_amdgcn_s_wait_asynccnt)
  __builtin_amdgcn_s_wait_asynccnt(0);
#else
  asm volatile("s_wait_asynccnt 0" ::: "memory");
#endif
  __syncthreads();

  // ---- Phase 2: build sparse table M[k][dl][i] = max v[i .. min(i+2^k-1, 63)] ----
  for (int k = 1; k < NLEV; ++k) {
    const int h = 1 << (k - 1);
    for (int e = tid; e < D_TILE * MAP_N; e += NTHR) {
      const int dl = e >> 6;
      const int i  = e & (MAP_N - 1);
      int i2 = i + h; i2 = (i2 > MAP_N - 1) ? (MAP_N - 1) : i2;
      M[k][dl][i] = fmaxf(M[k - 1][dl][i], M[k - 1][dl][i2]);
    }
    __syncthreads();
  }

  // ---- Phase 3: one wave per channel; half-wave per row, float4 per lane ----
  // Lanes 0-15 -> row r, lanes 16-31 -> row r+1; each lane emits 4 consecutive
  // columns as one b128 NT store. Rows r and r+1 are adjacent in memory, so each
  // wave iteration writes 512 contiguous bytes.
  const int dl   = tid >> 5;            // wave id 0..15 -> channel in tile
  const int lane = tid & 31;
  const int rsub = lane >> 4;           // 0 or 1: which row of the pair
  const int c0   = (lane & 15) << 2;    // 4 columns per lane
  float* obase = out + (size_t)(b * DMODEL + d0 + dl) * MAP_N * MAP_N;

  for (int r = 0; r < MAP_N; r += 2) {
    const int row = r + rsub;
    v4f pv;
    pv.x = tile_val(M, dl, row, c0);
    pv.y = tile_val(M, dl, row, c0 + 1);
    pv.z = tile_val(M, dl, row, c0 + 2);
    pv.w = tile_val(M, dl, row, c0 + 3);
    __builtin_nontemporal_store(pv, (v4f*)(obase + (size_t)row * MAP_N + c0));
  }
}

extern "C" void kernel_launch(void* const* d_in, const int* in_sizes, int n_in,
                              void* d_out, int out_size, void* d_ws, size_t ws_size,
                              hipStream_t stream) {
  const float* x = (const float*)d_in[0];
  float* out = (float*)d_out;
  const dim3 grid(BATCH * (DMODEL / D_TILE));   // 32 * 32 = 1024 workgroups
  const dim3 block(NTHR);                       // 16 wave32s
  gen2dmap_kernel<<<grid, block, 0, stream>>>(x, out);
}